// GraphRes_35064113005030
// MI455X (gfx1250) — compile-verified
//
#include <hip/hip_runtime.h>
#include <math.h>

namespace {

constexpr int   kN      = 200000;
constexpr int   kE      = 3200000;
constexpr int   kB      = 16;
constexpr int   kGrids  = 16;          // 4x4 voxel grid
constexpr int   kChOut  = 128;
constexpr int   kNumOut = 101;
constexpr float kVX     = 60.0f;       // 240/4
constexpr float kVY     = 45.0f;       // 180/4
constexpr float kEps    = 1e-5f;
constexpr int   kMTiles = kN / 16;     // 12500 exact

typedef float v2f __attribute__((ext_vector_type(2)));
typedef float v8f __attribute__((ext_vector_type(8)));

// 128-bit payload type for the gfx1250 async-to-LDS copy builtin
// (param 0 is AS1 "__device__" int4*, param 1 is the LDS-side pointer).
typedef int vint4 __attribute__((vector_size(16)));
typedef __attribute__((address_space(1))) vint4* as1_v4i;
typedef __attribute__((address_space(3))) vint4* as3_v4i;

__global__ void init_f32_kernel(float* __restrict__ p, float v, int n) {
  int i = blockIdx.x * blockDim.x + threadIdx.x;
  if (i < n) p[i] = v;
}

__global__ void deg_kernel(const int* __restrict__ ei, float* __restrict__ deg) {
  int e = blockIdx.x * blockDim.x + threadIdx.x;
  if (e < kE) atomicAdd(&deg[ei[kE + e]], 1.0f);
}

__global__ void rsqrt_kernel(const float* __restrict__ deg, float* __restrict__ dinv) {
  int i = blockIdx.x * blockDim.x + threadIdx.x;
  if (i < kN) dinv[i] = rsqrtf(deg[i]);
}

// Layer 1 GEMM: Cin == 1, trivial outer product.
__global__ void lin1_kernel(const float* __restrict__ x, const float* __restrict__ w,
                            float* __restrict__ out) {
  int i = blockIdx.x * blockDim.x + threadIdx.x;
  if (i < kN * 16) out[i] = x[i >> 4] * w[i & 15];
}

// Dense GEMM  C[N x COUT] = A[N x CIN] @ W[CIN x COUT]  via V_WMMA_F32_16X16X4_F32.
// Block = 256 threads = 8 waves; wave w owns M-tile blockIdx.x*8+w and the full
// COUT strip (COUT/16 accumulator tiles). W staged in LDS (<= 64KB) with the
// gfx1250 async copy engine (GLOBAL_LOAD_ASYNC_TO_LDS_B128, ASYNCcnt-tracked).
template <int CIN, int COUT>
__global__ __launch_bounds__(256) void gemm_wmma_kernel(const float* __restrict__ A,
                                                        const float* __restrict__ W,
                                                        float* __restrict__ C) {
  __shared__ float sW[CIN * COUT];

  // Async stage: 16 bytes per lane, no VGPR round-trip.
  constexpr int WBYTES = CIN * COUT * 4;
  {
    char* gsrc = const_cast<char*>((const char*)W);
    char* ldst = (char*)sW;
    for (int off = threadIdx.x * 16; off < WBYTES; off += 256 * 16) {
      __builtin_amdgcn_global_load_async_to_lds_b128(
          (as1_v4i)(gsrc + off), (as3_v4i)(ldst + off), 0, 0);
    }
    __builtin_amdgcn_s_wait_asynccnt(0);
  }
  __syncthreads();

  const int wave = threadIdx.x >> 5;
  const int lane = threadIdx.x & 31;
  const int mt   = blockIdx.x * 8 + wave;
  if (mt >= kMTiles) return;   // whole wave exits together -> EXEC all-1 for WMMA

  const int l15  = lane & 15;
  const int half = lane >> 4;
  constexpr int CT = COUT / 16;

  v8f acc[CT];
#pragma unroll
  for (int j = 0; j < CT; ++j)
#pragma unroll
    for (int v = 0; v < 8; ++v) acc[j][v] = 0.0f;

  // A frag (16x4 f32): lanes 0-15 supply K=k0,k0+1 of row M=l15; lanes 16-31 K=k0+2,k0+3.
  const float* arow = A + (size_t)(mt * 16 + l15) * CIN;
  for (int k0 = 0; k0 < CIN; k0 += 4) {
    const int kk = k0 + half * 2;
    v2f a;
    a.x = arow[kk];
    a.y = arow[kk + 1];
#pragma unroll
    for (int j = 0; j < CT; ++j) {
      const int col = j * 16 + l15;
      v2f b;                               // B frag (4x16): same K split across lane halves
      b.x = sW[kk * COUT + col];
      b.y = sW[(kk + 1) * COUT + col];
      acc[j] = __builtin_amdgcn_wmma_f32_16x16x4_f32(false, a, false, b, (short)0,
                                                     acc[j], false, false);
    }
  }

  // C/D layout: VGPR v holds M = v (lanes 0-15) / v+8 (lanes 16-31), N = l15.
  const int rbase = mt * 16 + half * 8;
#pragma unroll
  for (int j = 0; j < CT; ++j) {
    const int col = j * 16 + l15;
#pragma unroll
    for (int v = 0; v < 8; ++v) C[(size_t)(rbase + v) * COUT + col] = acc[j][v];
  }
}

// agg = hw * dinv[n]^2 + bias[c]   (self-loop term + GCN bias)
__global__ void self_bias_kernel(const float* __restrict__ hw, const float* __restrict__ dinv,
                                 const float* __restrict__ bias, float* __restrict__ agg,
                                 int cout, int total) {
  int i = blockIdx.x * blockDim.x + threadIdx.x;
  if (i >= total) return;
  int n = i / cout;
  int c = i - n * cout;
  float d = dinv[n];
  agg[i] = hw[i] * d * d + bias[c];
}

// agg[dst] += hw[src] * dinv[src]*dinv[dst] ; 4 channels per thread, L2-resident atomics.
__global__ void edge_agg_kernel(const int* __restrict__ ei, const float* __restrict__ dinv,
                                const float* __restrict__ hw, float* __restrict__ agg,
                                int cout4, int cout, int total) {
  int i = blockIdx.x * blockDim.x + threadIdx.x;
  if (i >= total) return;
  int e  = i / cout4;
  int c4 = (i - e * cout4) * 4;
  int src = ei[e];
  int dst = ei[kE + e];
  float w = dinv[src] * dinv[dst];
  const float4 v = *(const float4*)(hw + (size_t)src * cout + c4);
  float* o = agg + (size_t)dst * cout + c4;
  atomicAdd(o + 0, v.x * w);
  atomicAdd(o + 1, v.y * w);
  atomicAdd(o + 2, v.z * w);
  atomicAdd(o + 3, v.w * w);
}

// Column sums / sum-of-squares: register partials then float atomics on <=128 columns.
__global__ void bn_stats_kernel(const float* __restrict__ h, float* __restrict__ sums,
                                float* __restrict__ sumsq, int cout) {
  int gid = blockIdx.x * blockDim.x + threadIdx.x;
  int c = gid % cout;
  int r = gid / cout;
  int stride = (gridDim.x * blockDim.x) / cout;   // launch size divisible by cout
  float s = 0.f, s2 = 0.f;
  for (; r < kN; r += stride) {
    float v = h[(size_t)r * cout + c];
    s += v;
    s2 += v * v;
  }
  atomicAdd(&sums[c], s);
  atomicAdd(&sumsq[c], s2);
}

__global__ void bn_final_kernel(const float* __restrict__ sums, const float* __restrict__ sumsq,
                                const float* __restrict__ g, const float* __restrict__ beta,
                                float* __restrict__ scale, float* __restrict__ shift, int cout) {
  int c = threadIdx.x;
  if (c >= cout) return;
  float mean = sums[c] * (1.0f / (float)kN);
  float var  = sumsq[c] * (1.0f / (float)kN) - mean * mean;   // biased variance
  float sc   = g[c] * rsqrtf(var + kEps);
  scale[c] = sc;
  shift[c] = beta[c] - mean * sc;
}

// h = elu(h*scale + shift) (+ residual, added AFTER activation per reference)
__global__ void bn_act_kernel(float* __restrict__ h, const float* __restrict__ scale,
                              const float* __restrict__ shift, const float* __restrict__ res,
                              int cout, int total) {
  int i = blockIdx.x * blockDim.x + threadIdx.x;
  if (i >= total) return;
  int c = i % cout;
  float y = h[i] * scale[c] + shift[c];
  y = (y > 0.0f) ? y : (expf(y) - 1.0f);
  if (res) y += res[i];
  h[i] = y;
}

__device__ inline void atomic_max_f32(float* addr, float v) {
  if (v >= 0.0f)
    atomicMax((int*)addr, __float_as_int(v));          // nonneg: int order == float order
  else
    atomicMin((unsigned int*)addr, (unsigned int)__float_as_int(v));  // neg: reversed
}

__global__ void pool_kernel(const float* __restrict__ h, const float* __restrict__ pos,
                            const int* __restrict__ batch, float* __restrict__ pooled) {
  int i = blockIdx.x * blockDim.x + threadIdx.x;
  if (i >= kN * (kChOut / 4)) return;
  int n  = i / (kChOut / 4);
  int c4 = (i - n * (kChOut / 4)) * 4;
  int cx = (int)floorf(pos[n * 3 + 0] / kVX);
  int cy = (int)floorf(pos[n * 3 + 1] / kVY);
  cx = cx < 0 ? 0 : (cx > 3 ? 3 : cx);
  cy = cy < 0 ? 0 : (cy > 3 ? 3 : cy);
  int cluster = batch[n] * kGrids + cy * 4 + cx;
  const float4 v = *(const float4*)(h + (size_t)n * kChOut + c4);
  float* o = pooled + (size_t)cluster * kChOut + c4;
  atomic_max_f32(o + 0, v.x);
  atomic_max_f32(o + 1, v.y);
  atomic_max_f32(o + 2, v.z);
  atomic_max_f32(o + 3, v.w);
}

// out[b,o] = sum_k pooled[b,k] * fc_w[o,k]   (empty cells: -inf -> 0 at read)
__global__ void fc_kernel(const float* __restrict__ pooled, const float* __restrict__ fcw,
                          float* __restrict__ out) {
  int t = blockIdx.x * blockDim.x + threadIdx.x;
  if (t >= kB * kNumOut) return;
  int b = t / kNumOut;
  int o = t - b * kNumOut;
  const float* p = pooled + (size_t)b * (kGrids * kChOut);
  const float* w = fcw + (size_t)o * (kGrids * kChOut);
  float s = 0.f;
  for (int k = 0; k < kGrids * kChOut; ++k) {
    float pv = p[k];
    pv = (pv == -INFINITY) ? 0.0f : pv;
    s = fmaf(pv, w[k], s);
  }
  out[t] = s;
}

}  // namespace

extern "C" void kernel_launch(void* const* d_in, const int* in_sizes, int n_in,
                              void* d_out, int out_size, void* d_ws, size_t ws_size,
                              hipStream_t stream) {
  (void)in_sizes; (void)n_in; (void)out_size; (void)ws_size;

  // setup_inputs() insertion order: x, pos, {W,b,g,beta} x7, fc_w, edge_index, batch
  const float* x   = (const float*)d_in[0];
  const float* pos = (const float*)d_in[1];
  const float *W[8], *bias[8], *gam[8], *bet[8];
  for (int i = 1; i <= 7; ++i) {
    int b0 = 2 + 4 * (i - 1);
    W[i]    = (const float*)d_in[b0 + 0];
    bias[i] = (const float*)d_in[b0 + 1];
    gam[i]  = (const float*)d_in[b0 + 2];
    bet[i]  = (const float*)d_in[b0 + 3];
  }
  const float* fcw  = (const float*)d_in[30];
  const int* ei     = (const int*)d_in[31];
  const int* batchv = (const int*)d_in[32];

  // Workspace layout (floats).
  float* ws = (float*)d_ws;
  const size_t NB = (size_t)kN * 128;
  float* t0 = ws;            // GEMM output buffer
  float* t1 = t0 + NB;       // rotating activation buffers
  float* t2 = t1 + NB;
  float* t3 = t2 + NB;
  float* deg    = t3 + NB;
  float* dinv   = deg + kN;
  float* sums   = dinv + kN;
  float* sumsq  = sums + 128;
  float* scale  = sumsq + 128;
  float* shift  = scale + 128;
  float* pooled = shift + 128;   // kB * kGrids * kChOut

  const int TPB = 256;
  auto nblk = [](long long n) { return (unsigned)((n + 255) / 256); };

  // --- GCN normalization (self-loop-added degree) ---
  init_f32_kernel<<<nblk(kN), TPB, 0, stream>>>(deg, 1.0f, kN);
  deg_kernel<<<nblk(kE), TPB, 0, stream>>>(ei, deg);
  rsqrt_kernel<<<nblk(kN), TPB, 0, stream>>>(deg, dinv);

  // Shared post-GEMM chain: self+bias, edge scatter, BN stats/finalize, ELU(+residual).
  auto post = [&](float* hw, float* agg, int li, const float* res, int cout) {
    long long total = (long long)kN * cout;
    self_bias_kernel<<<nblk(total), TPB, 0, stream>>>(hw, dinv, bias[li], agg, cout, (int)total);
    long long et = (long long)kE * (cout / 4);
    edge_agg_kernel<<<nblk(et), TPB, 0, stream>>>(ei, dinv, hw, agg, cout / 4, cout, (int)et);
    init_f32_kernel<<<1, 256, 0, stream>>>(sums, 0.0f, 256);  // zeros sums AND sumsq
    bn_stats_kernel<<<1024, 256, 0, stream>>>(agg, sums, sumsq, cout);
    bn_final_kernel<<<1, 128, 0, stream>>>(sums, sumsq, gam[li], bet[li], scale, shift, cout);
    bn_act_kernel<<<nblk(total), TPB, 0, stream>>>(agg, scale, shift, res, cout, (int)total);
  };

  const unsigned GB = (unsigned)((kMTiles + 7) / 8);

  // L1 (1->16)
  lin1_kernel<<<nblk((long long)kN * 16), TPB, 0, stream>>>(x, W[1], t0);
  post(t0, t1, 1, nullptr, 16);
  // L2 (16->32)
  gemm_wmma_kernel<16, 32><<<GB, 256, 0, stream>>>(t1, W[2], t0);
  post(t0, t2, 2, nullptr, 32);           // t2 = sc
  // L3 (32->32)
  gemm_wmma_kernel<32, 32><<<GB, 256, 0, stream>>>(t2, W[3], t0);
  post(t0, t1, 3, nullptr, 32);
  // L4 (32->32) + residual sc(t2)
  gemm_wmma_kernel<32, 32><<<GB, 256, 0, stream>>>(t1, W[4], t0);
  post(t0, t3, 4, t2, 32);
  // L5 (32->128)
  gemm_wmma_kernel<32, 128><<<GB, 256, 0, stream>>>(t3, W[5], t0);
  post(t0, t1, 5, nullptr, 128);          // t1 = sc
  // L6 (128->128)
  gemm_wmma_kernel<128, 128><<<GB, 256, 0, stream>>>(t1, W[6], t0);
  post(t0, t2, 6, nullptr, 128);
  // L7 (128->128) + residual sc(t1)
  gemm_wmma_kernel<128, 128><<<GB, 256, 0, stream>>>(t2, W[7], t0);
  post(t0, t3, 7, t1, 128);

  // --- voxel-grid scatter-max pooling + FC ---
  init_f32_kernel<<<nblk(kB * kGrids * kChOut), TPB, 0, stream>>>(pooled, -INFINITY,
                                                                  kB * kGrids * kChOut);
  pool_kernel<<<nblk((long long)kN * (kChOut / 4)), TPB, 0, stream>>>(t3, pos, batchv, pooled);
  fc_kernel<<<nblk(kB * kNumOut), TPB, 0, stream>>>(pooled, fcw, (float*)d_out);
}